// Gaussian_loss_52879637348914
// MI455X (gfx1250) — compile-verified
//
#include <hip/hip_runtime.h>
#include <hip/hip_bf16.h>
#include <limits.h>

typedef float f2  __attribute__((ext_vector_type(2)));
typedef float f4  __attribute__((ext_vector_type(4)));
typedef int   i2  __attribute__((ext_vector_type(2)));
typedef float v2f __attribute__((ext_vector_type(2)));
typedef float v8f __attribute__((ext_vector_type(8)));

#define NBLK 1024
#define TPB  256
#define NQ   16   // padded quantity columns (6 real + 10 zero)

struct IPair { int a, b; };  // two smallest indices, ascending

__device__ __forceinline__ void ins2(IPair& p, int i) {
    if (i < p.a) { p.b = p.a; p.a = i; }
    else if (i < p.b) { p.b = i; }
}
__device__ __forceinline__ IPair merge2(IPair x, IPair y) {
    IPair r;
    if (x.a <= y.a) { r.a = x.a; r.b = min(x.b, y.a); }
    else            { r.a = y.a; r.b = min(y.b, x.a); }
    return r;
}

// ---------------------------------------------------------------------------
// Pass 1: bandwidth-bound streaming reduction. 1024 blocks x 256 threads.
// Two samples per iteration via b128 loads (4x b128 + 1x b64 per 72B/lane),
// NT hints (302MB streamed once > 192MB L2). Quantities:
// [0]=sum sq err, [1]=count(label==0), [2]=sum bt|label0, [3]=sum bi|label0,
// [4]=sum bt, [5]=sum bi. Plus per-block first-two indices of each class.
// ---------------------------------------------------------------------------
__global__ void gauss_loss_main_kernel(const f4* __restrict__ tx,
                                       const f4* __restrict__ im,
                                       const f4* __restrict__ pt,
                                       const f4* __restrict__ pi,
                                       const i2* __restrict__ lab,
                                       int n, float* __restrict__ Pf,
                                       int* __restrict__ Pi) {
    const int tid    = blockIdx.x * blockDim.x + threadIdx.x;
    const int stride = gridDim.x * blockDim.x;
    const int pairs  = n >> 1;

    float msum = 0.f, cnt0 = 0.f, s0t = 0.f, s0i = 0.f, stt = 0.f, sti = 0.f;
    IPair p0 = {INT_MAX, INT_MAX}, p1 = {INT_MAX, INT_MAX};

    auto accum = [&](float t0, float t1, float i0, float i1,
                     float q0, float q1, float r0, float r1, int L, int idx) {
        float bt  = logf(t0) - logf(t1);
        float bi  = logf(i0) - logf(i1);
        float bot = logf(q0) - logf(q1);
        float boi = logf(r0) - logf(r1);
        float c   = (L == 0) ? 3.0f : -3.0f;
        float dt  = bt - (bot + c);
        float di  = bi - (boi + c);
        msum += dt * dt + di * di;
        if (L == 0) { cnt0 += 1.0f; s0t += bt; s0i += bi; ins2(p0, idx); }
        else        { ins2(p1, idx); }
        stt += bt; sti += bi;
    };

    for (int j = tid; j < pairs; j += stride) {
        f4 tv = __builtin_nontemporal_load(&tx[j]);
        f4 iv = __builtin_nontemporal_load(&im[j]);
        f4 pv = __builtin_nontemporal_load(&pt[j]);
        f4 qv = __builtin_nontemporal_load(&pi[j]);
        i2 L2 = __builtin_nontemporal_load(&lab[j]);
        accum(tv.x, tv.y, iv.x, iv.y, pv.x, pv.y, qv.x, qv.y, L2.x, 2 * j);
        accum(tv.z, tv.w, iv.z, iv.w, pv.z, pv.w, qv.z, qv.w, L2.y, 2 * j + 1);
    }
    // odd-length tail (n is even for BS=2^23; kept for generality)
    if ((n & 1) && tid == 0) {
        const f2* txe = (const f2*)tx; const f2* ime = (const f2*)im;
        const f2* pte = (const f2*)pt; const f2* pie = (const f2*)pi;
        const int* le = (const int*)lab;
        int i = n - 1;
        f2 tv = txe[i], iv = ime[i], pv = pte[i], qv = pie[i];
        accum(tv.x, tv.y, iv.x, iv.y, pv.x, pv.y, qv.x, qv.y, le[i], i);
    }

    // wave32 butterfly reduction
    #pragma unroll
    for (int off = 16; off > 0; off >>= 1) {
        msum += __shfl_xor(msum, off, 32);
        cnt0 += __shfl_xor(cnt0, off, 32);
        s0t  += __shfl_xor(s0t,  off, 32);
        s0i  += __shfl_xor(s0i,  off, 32);
        stt  += __shfl_xor(stt,  off, 32);
        sti  += __shfl_xor(sti,  off, 32);
        IPair o0, o1;
        o0.a = __shfl_xor(p0.a, off, 32); o0.b = __shfl_xor(p0.b, off, 32);
        o1.a = __shfl_xor(p1.a, off, 32); o1.b = __shfl_xor(p1.b, off, 32);
        p0 = merge2(p0, o0); p1 = merge2(p1, o1);
    }

    __shared__ float sf[TPB / 32][6];
    __shared__ int   si[TPB / 32][4];
    const int wid = threadIdx.x >> 5, lane = threadIdx.x & 31;
    if (lane == 0) {
        sf[wid][0] = msum; sf[wid][1] = cnt0; sf[wid][2] = s0t;
        sf[wid][3] = s0i;  sf[wid][4] = stt;  sf[wid][5] = sti;
        si[wid][0] = p0.a; si[wid][1] = p0.b; si[wid][2] = p1.a; si[wid][3] = p1.b;
    }
    __syncthreads();
    if (threadIdx.x == 0) {
        float q[6] = {0.f, 0.f, 0.f, 0.f, 0.f, 0.f};
        IPair r0 = {INT_MAX, INT_MAX}, r1 = {INT_MAX, INT_MAX};
        #pragma unroll
        for (int w = 0; w < TPB / 32; ++w) {
            for (int k = 0; k < 6; ++k) q[k] += sf[w][k];
            IPair a = {si[w][0], si[w][1]}, b = {si[w][2], si[w][3]};
            r0 = merge2(r0, a); r1 = merge2(r1, b);
        }
        float* row = Pf + (size_t)blockIdx.x * NQ;
        for (int k = 0; k < 6; ++k)  row[k] = q[k];
        for (int k = 6; k < NQ; ++k) row[k] = 0.0f;
        int* irow = Pi + blockIdx.x * 4;
        irow[0] = r0.a; irow[1] = r0.b; irow[2] = r1.a; irow[3] = r1.b;
    }
}

// ---------------------------------------------------------------------------
// Pass 2: single wave. Column-reduce the [NBLK x 16] partials matrix as
// ones(16x4) x P via V_WMMA_F32_16X16X4_F32 (f32 in/out: same precision class
// as the reference's f32 sums). A = all-ones is layout-independent:
// D[m][n] = sum_k B[k][n]. Four accumulators break the D->C RAW chain.
// Then merge index pairs, recompute betanew for the 4 selected samples, and
// evaluate the erf tail on lane 0.
// ---------------------------------------------------------------------------
__global__ void gauss_loss_final_kernel(const float* __restrict__ Pf,
                                        const int* __restrict__ Pi,
                                        const f2* __restrict__ tx,
                                        const f2* __restrict__ im,
                                        const float* __restrict__ alpha,
                                        int n, float* __restrict__ out) {
    const int lane  = threadIdx.x;
    const int col   = lane & 15;
    const int half2 = (lane >> 4) * 2;  // lanes 0-15 -> K rows {0,1}; 16-31 -> {2,3}

    float sums[6];
#if defined(__has_builtin) && __has_builtin(__builtin_amdgcn_wmma_f32_16x16x4_f32)
    {
        v2f ones = {1.0f, 1.0f};
        v8f acc[4] = {};
        for (int r = 0; r < NBLK; r += 16) {
            #pragma unroll
            for (int s = 0; s < 4; ++s) {
                const int row = r + 4 * s + half2;
                v2f b;
                b.x = Pf[row * NQ + col];
                b.y = Pf[(row + 1) * NQ + col];
                acc[s] = __builtin_amdgcn_wmma_f32_16x16x4_f32(
                    false, ones, false, b, (short)0, acc[s], false, false);
            }
        }
        v8f d = (acc[0] + acc[1]) + (acc[2] + acc[3]);
        float mine = d[0];  // lane q (q<16) holds column-q total
        #pragma unroll
        for (int q = 0; q < 6; ++q) sums[q] = __shfl(mine, q, 32);
    }
#else
    {
        float mine = 0.0f;
        for (int r = 0; r < NBLK; ++r) mine += Pf[r * NQ + col];
        #pragma unroll
        for (int q = 0; q < 6; ++q) sums[q] = __shfl(mine, q, 32);
    }
#endif

    // merge first-two index pairs (min-2 merge is associative & commutative)
    IPair p0 = {INT_MAX, INT_MAX}, p1 = {INT_MAX, INT_MAX};
    for (int r = lane; r < NBLK; r += 32) {
        IPair a = {Pi[4 * r + 0], Pi[4 * r + 1]};
        IPair b = {Pi[4 * r + 2], Pi[4 * r + 3]};
        p0 = merge2(p0, a); p1 = merge2(p1, b);
    }
    #pragma unroll
    for (int off = 16; off > 0; off >>= 1) {
        IPair o0, o1;
        o0.a = __shfl_xor(p0.a, off, 32); o0.b = __shfl_xor(p0.b, off, 32);
        o1.a = __shfl_xor(p1.a, off, 32); o1.b = __shfl_xor(p1.b, off, 32);
        p0 = merge2(p0, o0); p1 = merge2(p1, o1);
    }

    if (lane == 0) {
        const float mse = sums[0] / (2.0f * (float)n);
        const float n0  = sums[1];
        const float n1  = (float)n - n0;
        const float m0t = sums[2] / n0,        m0i = sums[3] / n0;
        const float m1t = (sums[4] - sums[2]) / n1;
        const float m1i = (sums[5] - sums[3]) / n1;

        // betanew for the first two samples of each class
        auto bnew = [&](int i, float& bt, float& bi) {
            f2 tv = tx[i]; f2 iv = im[i];
            bt = logf(tv.x) - logf(tv.y);
            bi = logf(iv.x) - logf(iv.y);
        };
        float a0t, a0i, b0t, b0i, a1t, a1i, b1t, b1i;
        bnew(p0.a, a0t, a0i); bnew(p0.b, b0t, b0i);
        bnew(p1.a, a1t, a1i); bnew(p1.b, b1t, b1i);

        const float al = alpha[0], be = 1.0f - al;

        // cov_terms on 2-vectors: da = a - mean(a) => caa = (a0-a1)^2/2, etc.
        float u0 = a0t - a0i, v0 = b0t - b0i;
        float u1 = a1t - a1i, v1 = b1t - b1i;
        float c0aa = 0.5f * u0 * u0, c0bb = 0.5f * v0 * v0, c0ab = 0.5f * u0 * v0;
        float c1aa = 0.5f * u1 * u1, c1bb = 0.5f * v1 * v1, c1ab = 0.5f * u1 * v1;

        float num0 = al * m0t + be * m0i;
        float var0 = al * al * c0aa + be * be * c0bb + 2.0f * al * be * c0ab;
        float num1 = al * m1t + be * m1i;
        float var1 = al * al * c1aa + be * be * c1bb + 2.0f * al * be * c1ab;

        float g0 = num0 / (sqrtf(2.0f) * sqrtf(var0));
        float g1 = num1 / (sqrtf(2.0f) * sqrtf(var1));

        float score1 = (0.5f + 0.5f * erff(g0)) * 0.5f;
        float score2 = (0.5f - 0.5f * erff(g1)) * 0.5f;

        out[0] = 0.5f * mse - 0.5f * (score1 + score2);
    }
}

extern "C" void kernel_launch(void* const* d_in, const int* in_sizes, int n_in,
                              void* d_out, int out_size, void* d_ws, size_t ws_size,
                              hipStream_t stream) {
    const f4*    tx    = (const f4*)d_in[0];
    const f4*    im    = (const f4*)d_in[1];
    const f4*    pt    = (const f4*)d_in[2];
    const f4*    pi    = (const f4*)d_in[3];
    const float* alpha = (const float*)d_in[4];
    const i2*    lab   = (const i2*)d_in[5];
    const int    n     = in_sizes[5];  // BS (label count)

    float* Pf = (float*)d_ws;
    int*   Pi = (int*)((char*)d_ws + (size_t)NBLK * NQ * sizeof(float));

    gauss_loss_main_kernel<<<NBLK, TPB, 0, stream>>>(tx, im, pt, pi, lab, n, Pf, Pi);
    gauss_loss_final_kernel<<<1, 32, 0, stream>>>(Pf, Pi, (const f2*)d_in[0],
                                                  (const f2*)d_in[1], alpha, n,
                                                  (float*)d_out);
}